// FastFoodUpsample_24335284699546
// MI455X (gfx1250) — compile-verified
//
#include <hip/hip_runtime.h>
#include <hip/hip_bf16.h>

// FastFood upsample on MI455X (gfx1250).
//   w = FWHT_1024(z * BB[:1024])                          (first big FWHT collapses)
//   X[b][r][c] = w[b][Pi[r*1024+c] & 1023] * GG[r*1024+c]
//   Y = H1024 @ X @ H1024   (H_{2^20} = H_1024 (x) H_1024), keep rows r<961, scale.
// FWHT_1024 = (H16 (x) I64) . (I16 (x) H16 (x) I4) . (I256 (x) H4):
//   - H4: fused into registers at load time (or one LDS b128 pass after async copy)
//   - both H16 stages: V_WMMA_F32_16X16X4_F32 (exact fp32, +-1 operands from popcount)
// LDS tile is chunk-padded (64-float chunks at stride 68) to break bank conflicts.
// Bandwidth-bound: ~400 MB total traffic -> ~17us at 23.3 TB/s HBM; Pi/GG/w stay
// L2-resident (192 MB L2). Workspace: [ w: 32*1024 f32 | Tt: 32*2^20 f32 ] ~128.2 MiB.

typedef __attribute__((ext_vector_type(2))) float v2f;
typedef __attribute__((ext_vector_type(8))) float v8f;
typedef int v4i __attribute__((vector_size(16)));
typedef __attribute__((address_space(1))) v4i g_v4i;   // global int4 (builtin param 0)
typedef __attribute__((address_space(3))) v4i l_v4i;   // LDS int4    (builtin param 1)

#define NT   256
#define CH   68      // padded chunk stride (floats); 68*4 B = 272 = 17*16 -> b128 aligned
#define ROWS 1088    // padded row stride = 16*CH

#if defined(__has_builtin)
#  if __has_builtin(__builtin_amdgcn_global_load_async_to_lds_b128)
#    define FF_ASYNC 1
#  endif
#endif
#ifndef FF_ASYNC
#  define FF_ASYNC 0
#endif

__device__ __forceinline__ int ldsoff(int row, int c) {
    return row * ROWS + ((c >> 6) * CH) + (c & 63);
}

// H4 on 4 contiguous elements, fully in registers.
__device__ __forceinline__ float4 h4(float4 x) {
    float s0 = x.x + x.z, s2 = x.x - x.z;
    float s1 = x.y + x.w, s3 = x.y - x.w;
    return make_float4(s0 + s1, s0 - s1, s2 + s3, s2 - s3);
}

__device__ __forceinline__ void ff_wait_async() {
#if __has_builtin(__builtin_amdgcn_s_wait_asynccnt)
    __builtin_amdgcn_s_wait_asynccnt(0);
#else
    asm volatile("s_wait_asynccnt 0" ::: "memory");
#endif
}

// Both H16 stages of FWHT_1024 on a [16 rows][1024] chunk-padded LDS tile,
// via V_WMMA_F32_16X16X4_F32. 64 jobs per stage over 8 waves, disjoint regions.
__device__ __forceinline__ void fwht_wmma_stages(float* tile, int tid)
{
    const int wv   = tid >> 5;      // wave 0..7
    const int lane = tid & 31;
    const int m    = lane & 15;     // M (A) / N (B,D)
    const int hi   = lane >> 4;     // half-wave select
    const int kk0  = hi << 1;       // K pair base for A/B VGPR layout
    __syncthreads();
    // S2: H16 over k (stride 4 inside each 64-chunk). N tiles = 16 (i,j) combos.
    for (int job = wv; job < 64; job += 8) {
        const int row = job >> 2, t = job & 3;
        const int i2  = t * 4 + (m >> 2), j2 = m & 3;
        float* base = tile + row * ROWS + i2 * CH + j2;
        v8f acc = {0.f, 0.f, 0.f, 0.f, 0.f, 0.f, 0.f, 0.f};
        #pragma unroll
        for (int q = 0; q < 4; ++q) {
            const int ka = 4 * q + kk0;
            v2f a, bb;
            a.x = (__builtin_popcount(m & ka)       & 1) ? -1.0f : 1.0f;
            a.y = (__builtin_popcount(m & (ka + 1)) & 1) ? -1.0f : 1.0f;
            bb.x = base[ka * 4];
            bb.y = base[(ka + 1) * 4];
            acc = __builtin_amdgcn_wmma_f32_16x16x4_f32(
                false, a, false, bb, (short)0, acc, false, false);
        }
        #pragma unroll
        for (int v = 0; v < 8; ++v) base[(v + (hi << 3)) * 4] = acc[v];
    }
    __syncthreads();
    // S3: H16 over i (chunk index, stride CH). N tiles = 16 in-chunk columns.
    for (int job = wv; job < 64; job += 8) {
        const int row = job >> 2, colbase = (job & 3) << 4;
        float* base = tile + row * ROWS + colbase + m;
        v8f acc = {0.f, 0.f, 0.f, 0.f, 0.f, 0.f, 0.f, 0.f};
        #pragma unroll
        for (int q = 0; q < 4; ++q) {
            const int ka = 4 * q + kk0;
            v2f a, bb;
            a.x = (__builtin_popcount(m & ka)       & 1) ? -1.0f : 1.0f;
            a.y = (__builtin_popcount(m & (ka + 1)) & 1) ? -1.0f : 1.0f;
            bb.x = base[ka * CH];
            bb.y = base[(ka + 1) * CH];
            acc = __builtin_amdgcn_wmma_f32_16x16x4_f32(
                false, a, false, bb, (short)0, acc, false, false);
        }
        #pragma unroll
        for (int v = 0; v < 8; ++v) base[(v + (hi << 3)) * CH] = acc[v];
    }
    __syncthreads();
}

// Kernel 1: w[b][:] = FWHT_1024(z[b][:] * BB[:1024]); 2 blocks x 16 batch rows.
__global__ void ff_seed(const float* __restrict__ z, const float* __restrict__ BBv,
                        float* __restrict__ wout)
{
    extern __shared__ float tile[];
    const int tid = threadIdx.x;
    const int b0  = blockIdx.x * 16;
    for (int g = tid; g < 4096; g += NT) {
        const int e = g << 2, row = e >> 10, c = e & 1023;
        float4 zz = *(const float4*)(z + b0 * 1024 + e);
        float4 bv = *(const float4*)(BBv + c);
        *(float4*)(tile + ldsoff(row, c)) =
            h4(make_float4(zz.x * bv.x, zz.y * bv.y, zz.z * bv.z, zz.w * bv.w));
    }
    fwht_wmma_stages(tile, tid);
    for (int g = tid; g < 4096; g += NT) {
        const int e = g << 2;
        *(float4*)(wout + b0 * 1024 + e) = *(float4*)(tile + ldsoff(e >> 10, e & 1023));
    }
}

// Kernel 2: c-direction FWHT of X rows r0..r0+15 for batch b; transposed store
// Tt[b][c][r] so the r-pass reads contiguously (64 B per 16-r span).
__global__ void ff_rows(const float* __restrict__ w, const float* __restrict__ GG,
                        const int* __restrict__ Pi, float* __restrict__ Tt)
{
    extern __shared__ float tile[];
    const int tid = threadIdx.x;
    const int b   = blockIdx.y;              // 0..31
    const int rt  = blockIdx.x;              // 0..63 (row-tile)
    const float* wb = w + b * 1024;
    const size_t base = (size_t)rt * 16384;
    for (int g = tid; g < 4096; g += NT) {
        const int e = g << 2, row = e >> 10, c = e & 1023;
        int4   p4 = *(const int4*)(Pi + base + e);
        float4 g4 = *(const float4*)(GG + base + e);
        float4 x  = make_float4(wb[p4.x & 1023] * g4.x, wb[p4.y & 1023] * g4.y,
                                wb[p4.z & 1023] * g4.z, wb[p4.w & 1023] * g4.w);
        *(float4*)(tile + ldsoff(row, c)) = h4(x);
    }
    fwht_wmma_stages(tile, tid);
    float* dst = Tt + ((size_t)b << 20) + rt * 16;
    for (int c = tid; c < 1024; c += NT) {
        const int roff = ((c >> 6) * CH) + (c & 63);
        float buf[16];
        #pragma unroll
        for (int j = 0; j < 16; ++j) buf[j] = tile[j * ROWS + roff];
        float4* o = (float4*)(dst + (size_t)c * 1024);
        o[0] = make_float4(buf[0],  buf[1],  buf[2],  buf[3]);
        o[1] = make_float4(buf[4],  buf[5],  buf[6],  buf[7]);
        o[2] = make_float4(buf[8],  buf[9],  buf[10], buf[11]);
        o[3] = make_float4(buf[12], buf[13], buf[14], buf[15]);
    }
}

// Kernel 3: r-direction FWHT for 16 c's of batch b, scale, scatter to the
// block-concatenated output layout. Input is a contiguous 64 KB block ->
// async copy straight into the padded LDS tile (ASYNCcnt path), H4 in one
// LDS b128 pass, then the two WMMA stages.
__global__ void ff_cols(const float* __restrict__ Tt, const float* __restrict__ divisor,
                        float* __restrict__ out)
{
    extern __shared__ float tile[];
    const int tid = threadIdx.x;
    const int b   = blockIdx.y;              // 0..31
    const int ct  = blockIdx.x;              // 0..63 (col-tile)
    const float* src = Tt + ((size_t)b << 20) + (size_t)ct * 16384;
#if FF_ASYNC
    for (int g = tid; g < 4096; g += NT) {
        const int e = g << 2, cc = e >> 10, r = e & 1023;
        __builtin_amdgcn_global_load_async_to_lds_b128(
            (g_v4i*)(src + e),
            (l_v4i*)(tile + ldsoff(cc, r)), 0, 0);
    }
    ff_wait_async();
#else
    for (int g = tid; g < 4096; g += NT) {
        const int e = g << 2;
        *(float4*)(tile + ldsoff(e >> 10, e & 1023)) = *(const float4*)(src + e);
    }
#endif
    __syncthreads();
    for (int g = tid; g < 4096; g += NT) {   // H4 pass, b128 in LDS
        const int e = g << 2;
        float4* p = (float4*)(tile + ldsoff(e >> 10, e & 1023));
        *p = h4(*p);
    }
    fwht_wmma_stages(tile, tid);
    const float inv = 1.0f / (divisor[0] * sqrtf(984064.0f / 1048576.0f));
    const int c0 = ct * 16;
    for (int r = tid; r < 1024; r += NT) {
        if (r >= 961) continue;              // k >= TOTAL_DIM dropped
        const size_t k0 = (size_t)r * 1024 + c0;
        size_t idx;
        if      (k0 < 786432) idx = (size_t)b * 786432 + k0;
        else if (k0 < 787200) idx = 25165824UL + (size_t)b * 768    + (k0 - 786432);
        else if (k0 < 983808) idx = 25190400UL + (size_t)b * 196608 + (k0 - 787200);
        else                  idx = 31481856UL + (size_t)b * 256    + (k0 - 983808);
        const int roff = ((r >> 6) * CH) + (r & 63);
        float buf[16];
        #pragma unroll
        for (int j = 0; j < 16; ++j) buf[j] = tile[j * ROWS + roff] * inv;
        float4* o = (float4*)(out + idx);    // idx is 16-float aligned
        o[0] = make_float4(buf[0],  buf[1],  buf[2],  buf[3]);
        o[1] = make_float4(buf[4],  buf[5],  buf[6],  buf[7]);
        o[2] = make_float4(buf[8],  buf[9],  buf[10], buf[11]);
        o[3] = make_float4(buf[12], buf[13], buf[14], buf[15]);
    }
}

extern "C" void kernel_launch(void* const* d_in, const int* in_sizes, int n_in,
                              void* d_out, int out_size, void* d_ws, size_t ws_size,
                              hipStream_t stream) {
    const float* z       = (const float*)d_in[0];   // [32,1024]
    const float* BBv     = (const float*)d_in[1];   // [2^20] (only first 1024 used)
    const float* GG      = (const float*)d_in[2];   // [2^20]
    const float* divisor = (const float*)d_in[3];   // [1]
    const int*   Pi      = (const int*)  d_in[4];   // [2^20]
    float* out = (float*)d_out;                     // 32*984064 f32

    float* w  = (float*)d_ws;                       // 32*1024 f32
    float* Tt = w + 32 * 1024;                      // 32*2^20 f32

    const size_t lds = 16 * ROWS * sizeof(float);   // 69632 B padded tile
    ff_seed<<<2,            NT, lds, stream>>>(z, BBv, w);
    ff_rows<<<dim3(64, 32), NT, lds, stream>>>(w, GG, Pi, Tt);
    ff_cols<<<dim3(64, 32), NT, lds, stream>>>(Tt, divisor, out);
}